// GATRanker_14448269983837
// MI455X (gfx1250) — compile-verified
//
#include <hip/hip_runtime.h>
#include <hip/hip_bf16.h>
#include <cfloat>

typedef __attribute__((ext_vector_type(2))) float v2f;
typedef __attribute__((ext_vector_type(8))) float v8f;

#define IN_CH 128
#define HID   64
#define NEG_SLOPE 0.2f

// ---- order-preserving float<->uint key for atomicMax on floats ----
__device__ __forceinline__ unsigned floatKey(float f) {
    unsigned b = __float_as_uint(f);
    return (b & 0x80000000u) ? ~b : (b | 0x80000000u);
}
__device__ __forceinline__ float keyFloat(unsigned k) {
    unsigned b = (k & 0x80000000u) ? (k & 0x7FFFFFFFu) : ~k;
    return __uint_as_float(b);
}
__device__ __forceinline__ float leaky(float v) {
    return v > 0.0f ? v : NEG_SLOPE * v;
}

// ================= K1: h = x @ W via V_WMMA_F32_16X16X4_F32 =================
// One wave computes one 16-row tile of h across all 64 output columns
// (4 accumulators of 16 cols each). 32 k-steps of 4 -> 128 WMMAs/wave.
__global__ void gat_gemm_wmma(const float* __restrict__ x,
                              const float* __restrict__ W,
                              float* __restrict__ h,
                              int nTiles) {
    int wave = blockIdx.x * (blockDim.x >> 5) + (threadIdx.x >> 5);
    if (wave >= nTiles) return;              // wave-uniform: EXEC stays all-ones
    int lane  = threadIdx.x & 31;
    int m0    = wave << 4;                   // first row of this tile
    int half  = lane >> 4;                   // 0 for lanes 0-15, 1 for 16-31
    int l15   = lane & 15;

    // A fragment: lane l15 holds row m0+l15; VGPR0=K(+0 or +2), VGPR1=K(+1 or +3)
    const float* xr = x + (size_t)(m0 + l15) * IN_CH + (half << 1);
    // B fragment: lane l15 holds col n0+l15; VGPR0=rowK(+0/+2), VGPR1=rowK(+1/+3)
    const float* wr = W + (size_t)(half << 1) * HID + l15;

    v8f acc0 = {}, acc1 = {}, acc2 = {}, acc3 = {};

    for (int k0 = 0; k0 < IN_CH; k0 += 4) {
        v2f a;
        a[0] = xr[k0];
        a[1] = xr[k0 + 1];
        const float* w0 = wr + (size_t)k0 * HID;       // K row (k0 + 2*half)
        const float* w1 = w0 + HID;                    // K row (k0 + 2*half + 1)
        v2f b0, b1, b2, b3;
        b0[0] = w0[0];  b0[1] = w1[0];
        b1[0] = w0[16]; b1[1] = w1[16];
        b2[0] = w0[32]; b2[1] = w1[32];
        b3[0] = w0[48]; b3[1] = w1[48];
        acc0 = __builtin_amdgcn_wmma_f32_16x16x4_f32(false, a, false, b0, (short)0, acc0, false, false);
        acc1 = __builtin_amdgcn_wmma_f32_16x16x4_f32(false, a, false, b1, (short)0, acc1, false, false);
        acc2 = __builtin_amdgcn_wmma_f32_16x16x4_f32(false, a, false, b2, (short)0, acc2, false, false);
        acc3 = __builtin_amdgcn_wmma_f32_16x16x4_f32(false, a, false, b3, (short)0, acc3, false, false);
    }

    // C/D layout: VGPR i -> row m0+i (lanes 0-15) or m0+8+i (lanes 16-31), col = lane&15
#pragma unroll
    for (int i = 0; i < 8; ++i) {
        int row = m0 + i + (half << 3);
        float* hr = h + (size_t)row * HID + l15;
        hr[0]  = acc0[i];
        hr[16] = acc1[i];
        hr[32] = acc2[i];
        hr[48] = acc3[i];
    }
}

// ============ K2: per-node attention scalars + state init ============
// One wave per node: a_src[n]=h[n]*att_src, a_dst[n]=h[n]*att_dst,
// out row = bias, max-key = 0 (== -inf), denom = 0.
__global__ void gat_node_init(const float* __restrict__ h,
                              const float* __restrict__ att_src,
                              const float* __restrict__ att_dst,
                              const float* __restrict__ bias,
                              float* __restrict__ a_src,
                              float* __restrict__ a_dst,
                              unsigned* __restrict__ mkey,
                              float* __restrict__ denom,
                              float* __restrict__ out,
                              int N) {
    int node = blockIdx.x * (blockDim.x >> 5) + (threadIdx.x >> 5);
    if (node >= N) return;
    int lane = threadIdx.x & 31;

    const float* hr = h + (size_t)node * HID;
    float h0 = hr[lane], h1 = hr[lane + 32];
    float ps = h0 * att_src[lane] + h1 * att_src[lane + 32];
    float pd = h0 * att_dst[lane] + h1 * att_dst[lane + 32];
#pragma unroll
    for (int off = 16; off; off >>= 1) {
        ps += __shfl_xor(ps, off, 32);
        pd += __shfl_xor(pd, off, 32);
    }
    float* orow = out + (size_t)node * HID;
    orow[lane]      = bias[lane];
    orow[lane + 32] = bias[lane + 32];
    if (lane == 0) {
        a_src[node] = ps;
        a_dst[node] = pd;
        mkey[node]  = 0u;      // smaller than key of any finite float
        denom[node] = 0.0f;
    }
}

// ============ K3: segment max of leaky-relu logits ============
__global__ void gat_edge_max(const int* __restrict__ ei,
                             const float* __restrict__ a_src,
                             const float* __restrict__ a_dst,
                             unsigned* __restrict__ mkey,
                             int E, int total) {
    int i = blockIdx.x * blockDim.x + threadIdx.x;
    if (i >= total) return;
    int s, d;
    if (i < E) { s = ei[i]; d = ei[E + i]; } else { s = d = i - E; }
    float e = leaky(a_src[s] + a_dst[d]);
    atomicMax(&mkey[d], floatKey(e));
}

// ============ K4: w = exp(e - m[dst]); denom[dst] += w ============
__global__ void gat_edge_exp(const int* __restrict__ ei,
                             const float* __restrict__ a_src,
                             const float* __restrict__ a_dst,
                             const unsigned* __restrict__ mkey,
                             float* __restrict__ denom,
                             float* __restrict__ alpha,
                             int E, int total) {
    int i = blockIdx.x * blockDim.x + threadIdx.x;
    if (i >= total) return;
    int s, d;
    if (i < E) { s = ei[i]; d = ei[E + i]; } else { s = d = i - E; }
    float e = leaky(a_src[s] + a_dst[d]);
    float w = __expf(e - keyFloat(mkey[d]));
    alpha[i] = w;                     // unnormalized, fixed up in next pass
    atomicAdd(&denom[d], w);
}

// ============ K5: normalize alpha in place ============
__global__ void gat_edge_norm(const int* __restrict__ ei,
                              const float* __restrict__ denom,
                              float* __restrict__ alpha,
                              int E, int total) {
    int i = blockIdx.x * blockDim.x + threadIdx.x;
    if (i >= total) return;
    int d = (i < E) ? ei[E + i] : (i - E);
    alpha[i] = alpha[i] / (denom[d] + 1e-16f);
}

// ============ K6: out[dst] += alpha * h[src], 64 threads per edge ============
__global__ void gat_edge_agg(const int* __restrict__ ei,
                             const float* __restrict__ h,
                             const float* __restrict__ alpha,
                             float* __restrict__ out,
                             int E, int total) {
    size_t t = (size_t)blockIdx.x * blockDim.x + threadIdx.x;
    int i = (int)(t >> 6);
    if (i >= total) return;
    int c = (int)(t & 63);
    int s, d;
    if (i < E) { s = ei[i]; d = ei[E + i]; } else { s = d = i - E; }
    float a = alpha[i];
    atomicAdd(&out[(size_t)d * HID + c], a * h[(size_t)s * HID + c]);
}

extern "C" void kernel_launch(void* const* d_in, const int* in_sizes, int n_in,
                              void* d_out, int out_size, void* d_ws, size_t ws_size,
                              hipStream_t stream) {
    const float* x       = (const float*)d_in[0];
    const int*   ei      = (const int*)d_in[1];
    const float* W       = (const float*)d_in[2];
    const float* att_src = (const float*)d_in[3];
    const float* att_dst = (const float*)d_in[4];
    const float* bias    = (const float*)d_in[5];

    const int N     = in_sizes[0] / IN_CH;   // 100000
    const int E     = in_sizes[1] / 2;       // 1600000
    const int total = E + N;

    float* out   = (float*)d_out;                         // [N, 64]
    float* alpha = (float*)d_out + (size_t)N * HID;       // [E+N]

    // workspace layout
    float*    h     = (float*)d_ws;                       // N*64
    float*    a_src = h + (size_t)N * HID;                // N
    float*    a_dst = a_src + N;                          // N
    unsigned* mkey  = (unsigned*)(a_dst + N);             // N
    float*    denom = (float*)(mkey + N);                 // N

    // K1: GEMM (WMMA f32 16x16x4), 8 waves/block, one 16-row tile per wave
    int nTiles = N / 16;
    gat_gemm_wmma<<<(nTiles + 7) / 8, 256, 0, stream>>>(x, W, h, nTiles);

    // K2: node scalars + init (one wave per node)
    gat_node_init<<<(N + 7) / 8, 256, 0, stream>>>(h, att_src, att_dst, bias,
                                                   a_src, a_dst, mkey, denom, out, N);

    const int TB = 256;
    gat_edge_max <<<(total + TB - 1) / TB, TB, 0, stream>>>(ei, a_src, a_dst, mkey, E, total);
    gat_edge_exp <<<(total + TB - 1) / TB, TB, 0, stream>>>(ei, a_src, a_dst, mkey, denom, alpha, E, total);
    gat_edge_norm<<<(total + TB - 1) / TB, TB, 0, stream>>>(ei, denom, alpha, E, total);

    size_t aggThreads = (size_t)total * HID;
    gat_edge_agg <<<(unsigned)((aggThreads + TB - 1) / TB), TB, 0, stream>>>(ei, h, alpha, out, E, total);
}